// ChamferDistance_14714557956150
// MI455X (gfx1250) — compile-verified
//
#include <hip/hip_runtime.h>

typedef __attribute__((ext_vector_type(2))) float v2f;
typedef __attribute__((ext_vector_type(8))) float v8f;

#define B_ 4
#define N_ 8192
#define NT_ (N_ / 16)          // 512 16-wide tiles per side
#define WAVES_PER_WG 8
#define WG_THREADS (WAVES_PER_WG * 32)
#define MGROUPS (NT_ / WAVES_PER_WG)   // 64 workgroups per batch

// ---------------------------------------------------------------------------
// Prep: fuse coords + registration, pack WMMA-friendly point data.
//   ptsA[i] = {gx, gy, gz, 1}            (gt points, A rows)
//   ptsB[j] = {-2qx, -2qy, -2qz, |q|^2}  (gen points, B cols)
//   nA[i]   = |p|^2                      (added in epilogue per row)
//   gcol[j] = FLT_MAX bits               (global col-min init)
// ---------------------------------------------------------------------------
__global__ void chamfer_prep(const float* __restrict__ pred,
                             const float* __restrict__ gt,
                             const float* __restrict__ coords,
                             float4* __restrict__ ptsA,
                             float4* __restrict__ ptsB,
                             float* __restrict__ nA,
                             unsigned* __restrict__ gcol) {
    int idx = blockIdx.x * blockDim.x + threadIdx.x;
    if (idx >= B_ * N_) return;
    int b = idx / N_;
    int i = idx - b * N_;
    const size_t base = (size_t)b * 3 * N_ + i;

    float cx = coords[base], cy = coords[base + N_], cz = coords[base + 2 * N_];
    float gx = cx + gt[base], gy = cy + gt[base + N_], gz = cz + gt[base + 2 * N_];
    float qx = cx + pred[base], qy = cy + pred[base + N_], qz = cz + pred[base + 2 * N_];

    ptsA[idx] = make_float4(gx, gy, gz, 1.0f);
    nA[idx]   = gx * gx + gy * gy + gz * gz;
    ptsB[idx] = make_float4(-2.0f * qx, -2.0f * qy, -2.0f * qz,
                            qx * qx + qy * qy + qz * qz);
    gcol[idx] = 0x7F7FFFFFu;  // FLT_MAX bits
}

// ---------------------------------------------------------------------------
// Main tiles: one wave owns one 16-row m-tile (gt rows), loops over all 512
// n-tiles (gen cols), 4 tiles per iteration. One V_WMMA_F32_16X16X4_F32 per
// 16x16 distance tile; 4 B-fragment loads in flight per iteration.
// ---------------------------------------------------------------------------
__global__ __launch_bounds__(WG_THREADS)
void chamfer_tiles(const float4* __restrict__ ptsA,
                   const float4* __restrict__ ptsB,
                   const float* __restrict__ nA,
                   unsigned* __restrict__ gcol,
                   float* __restrict__ rowsum) {
    __shared__ unsigned cmin_lds[N_];  // 32 KB: per-block col-min (uint = float bits)

    const int b    = blockIdx.x / MGROUPS;
    const int mg   = blockIdx.x - b * MGROUPS;
    const int wave = threadIdx.x >> 5;
    const int lane = threadIdx.x & 31;
    const int half = lane >> 4;        // 0: K={0,1}, 1: K={2,3}
    const int l16  = lane & 15;
    const int mtile = mg * WAVES_PER_WG + wave;

    for (int i = threadIdx.x; i < N_; i += WG_THREADS) cmin_lds[i] = 0x7F7FFFFFu;
    __syncthreads();

    // A fragment (loaded once): lane's float2 of point row mtile*16 + l16.
    const float2* pa = (const float2*)&ptsA[(size_t)b * N_ + mtile * 16 + l16];
    float2 av = pa[half];
    v2f afrag;
    afrag.x = av.x;
    afrag.y = av.y;

    // |p|^2 for the 8 accumulator rows this lane owns (rows half*8 + v).
    const float* nabase = &nA[(size_t)b * N_ + mtile * 16 + half * 8];
    float nar[8];
#pragma unroll
    for (int v = 0; v < 8; ++v) nar[v] = nabase[v];

    float rowmin[8];
#pragma unroll
    for (int v = 0; v < 8; ++v) rowmin[v] = 3.0e38f;

    // Per-lane B-fragment base: &ptsB[b*N + l16] + half*8 bytes; each tile
    // advances 16 points = 256 bytes; 4-tile group advances 1024 bytes.
    const char* p = (const char*)ptsB + ((size_t)b * N_ + l16) * 16 + half * 8;

    for (int nt = 0; nt < NT_; nt += 4, p += 1024) {
        float2 bq0 = *(const float2*)(p);
        float2 bq1 = *(const float2*)(p + 256);
        float2 bq2 = *(const float2*)(p + 512);
        float2 bq3 = *(const float2*)(p + 768);
        float2 bq[4] = {bq0, bq1, bq2, bq3};

#pragma unroll
        for (int u = 0; u < 4; ++u) {
            v2f bfrag;
            bfrag.x = bq[u].x;
            bfrag.y = bq[u].y;

            v8f c = {};
            // D = A(16x4) * B(4x16): yields -2*p.q + |q|^2 per element.
            c = __builtin_amdgcn_wmma_f32_16x16x4_f32(
                    false, afrag, false, bfrag, (short)0, c, false, false);

            float cmin = 3.0e38f;
#pragma unroll
            for (int v = 0; v < 8; ++v) {
                float d = fmaxf(c[v] + nar[v], 0.0f);  // full squared distance
                rowmin[v] = fminf(rowmin[v], d);
                cmin = fminf(cmin, d);
            }
            // All 32 lanes merge their 8-row partial min into the column slot;
            // the LDS atomic performs the cross-half combine (no shuffles).
            atomicMin(&cmin_lds[(nt + u) * 16 + l16], __float_as_uint(cmin));
        }
    }

    // Row-min reduction across the 16 columns each half-lane-group covers,
    // then sum the 16 row minima of this m-tile (dist1 partial).
    float rs = 0.0f;
#pragma unroll
    for (int v = 0; v < 8; ++v) {
        float r = rowmin[v];
        r = fminf(r, __shfl_xor(r, 1, 32));
        r = fminf(r, __shfl_xor(r, 2, 32));
        r = fminf(r, __shfl_xor(r, 4, 32));
        r = fminf(r, __shfl_xor(r, 8, 32));
        rs += r;
    }
    rs += __shfl_xor(rs, 16, 32);
    if (lane == 0) rowsum[(size_t)b * NT_ + mtile] = rs;

    __syncthreads();
    // Flush block-local col mins into the global per-batch col-min array.
    for (int i = threadIdx.x; i < N_; i += WG_THREADS)
        atomicMin(&gcol[(size_t)b * N_ + i], cmin_lds[i]);
}

// ---------------------------------------------------------------------------
// Final deterministic reduction: sum row partials + col mins, scale by 1/B.
// ---------------------------------------------------------------------------
__global__ void chamfer_final(const float* __restrict__ rowsum,
                              const unsigned* __restrict__ gcol,
                              float* __restrict__ out) {
    __shared__ float sdata[256];
    float s = 0.0f;
    for (int i = threadIdx.x; i < B_ * NT_; i += 256) s += rowsum[i];
    for (int i = threadIdx.x; i < B_ * N_; i += 256) s += __uint_as_float(gcol[i]);
    sdata[threadIdx.x] = s;
    __syncthreads();
    for (int off = 128; off > 0; off >>= 1) {
        if ((int)threadIdx.x < off) sdata[threadIdx.x] += sdata[threadIdx.x + off];
        __syncthreads();
    }
    if (threadIdx.x == 0) out[0] = sdata[0] / (float)B_;
}

extern "C" void kernel_launch(void* const* d_in, const int* in_sizes, int n_in,
                              void* d_out, int out_size, void* d_ws, size_t ws_size,
                              hipStream_t stream) {
    const float* pred   = (const float*)d_in[0];  // registration_pred [B,3,N]
    const float* gt     = (const float*)d_in[1];  // registration_gt   [B,3,N]
    const float* coords = (const float*)d_in[2];  // coords            [B,3,N]
    float* out = (float*)d_out;

    char* ws = (char*)d_ws;
    float4*   ptsA   = (float4*)(ws);                              // 512 KB
    float4*   ptsB   = (float4*)(ws + (size_t)B_ * N_ * 16);       // 512 KB
    float*    nA     = (float*)(ws + (size_t)B_ * N_ * 32);        // 128 KB
    unsigned* gcol   = (unsigned*)(ws + (size_t)B_ * N_ * 36);     // 128 KB
    float*    rowsum = (float*)(ws + (size_t)B_ * N_ * 40);        //   8 KB

    chamfer_prep<<<(B_ * N_ + 255) / 256, 256, 0, stream>>>(
        pred, gt, coords, ptsA, ptsB, nA, gcol);

    chamfer_tiles<<<B_ * MGROUPS, WG_THREADS, 0, stream>>>(
        ptsA, ptsB, nA, gcol, rowsum);

    chamfer_final<<<1, 256, 0, stream>>>(rowsum, gcol, out);
}